// CNNGCN_18236431139458
// MI455X (gfx1250) — compile-verified
//
#include <hip/hip_runtime.h>
#include <hip/hip_bf16.h>
#include <stdint.h>

// ---------------- problem constants ----------------
#define Bn   32
#define Hn   2050
#define En   128
#define Nn   2048
#define Cn   128
#define NEn  32768
#define MLPn 100
#define KCONV 384            // 3*128
#define MROWS (Bn * Nn)      // 65536
#define HNC   (Bn * Nn * Cn) // 8388608 floats

typedef __attribute__((ext_vector_type(2))) float v2f;
typedef __attribute__((ext_vector_type(8))) float v8f;
typedef __attribute__((ext_vector_type(4))) int   v4i;

// ---------------- async-copy availability probe ----------------
#if defined(__HIP_DEVICE_COMPILE__) && defined(__gfx1250__) && __has_builtin(__builtin_amdgcn_global_load_async_to_lds_b128)
#define USE_ASYNC 1
#else
#define USE_ASYNC 0
#endif

#if USE_ASYNC
// Parameter types per hipcc diagnostic: (v4i AS1*, v4i AS3*, imm int, imm int).
// Casting through uintptr_t sidesteps generic->numbered-AS conversion rules;
// low 32 bits of a flat LDS pointer are the AS3 offset on AMDGPU.
typedef v4i __attribute__((address_space(1))) gv4i;
typedef v4i __attribute__((address_space(3))) lv4i;

__device__ __forceinline__ void async_copy16(const float* g, float* l) {
  __builtin_amdgcn_global_load_async_to_lds_b128(
      (gv4i*)(uintptr_t)g, (lv4i*)(uint32_t)(uintptr_t)l, 0, 0);
}
__device__ __forceinline__ void async_wait0() {
#if __has_builtin(__builtin_amdgcn_s_wait_asynccnt)
  __builtin_amdgcn_s_wait_asynccnt(0);
#else
  asm volatile("s_wait_asynccnt 0" ::: "memory");
#endif
}
#endif

// ---------------- small helper kernels ----------------
__global__ void k_init_cgx(float* deg, float* feat, const float* b1) {
  int i = blockIdx.x * blockDim.x + threadIdx.x;
  if (i < Nn) deg[i] = 0.0f;
  if (i < Bn * MLPn) feat[i] = b1[i % MLPn];
}

__global__ void k_deg_cgx(const long long* ei, float* deg) {
  int e = blockIdx.x * blockDim.x + threadIdx.x;
  if (e < NEn) {
    int d = (int)ei[NEn + e];
    atomicAdd(&deg[d], 1.0f);
  }
}

__global__ void k_dinv_cgx(float* deg) {
  int i = blockIdx.x * blockDim.x + threadIdx.x;
  if (i < Nn) deg[i] = rsqrtf(deg[i] + 1.0f); // +1 self-loop; deg>=1 always
}

__global__ void k_norm_cgx(const long long* ei, const float* dinv, float* norm) {
  int e = blockIdx.x * blockDim.x + threadIdx.x;
  if (e < NEn) {
    int s = (int)ei[e];
    int d = (int)ei[NEn + e];
    norm[e] = dinv[s] * dinv[d];
  }
}

// ---------------- conv-as-GEMM with f32 WMMA ----------------
// M=65536 (b,n), K=384 (contiguous in x per row), N=128 (out channels)
#define BM 64
#define BN 32
#define BK 64
#define LDA 68
#define LDB 33

__global__ __launch_bounds__(256) void k_conv_gemm_cgx(const float* __restrict__ x,
                                                       const float* __restrict__ cw,
                                                       const float* __restrict__ cb,
                                                       float* __restrict__ out) {
  __shared__ float sA[BM * LDA];
  __shared__ float sB[BK * LDB];
  const int tid  = threadIdx.x;
  const int M0   = blockIdx.x * BM;
  const int N0   = blockIdx.y * BN;
  const int lane = tid & 31;
  const int wave = tid >> 5;
  const int wm   = wave >> 1;   // 0..3
  const int wn   = wave & 1;    // 0..1

  // A staging: r = row in tile, q selects 16-float segment of the 64-float chunk
  const int r = tid & 63;
  const int q = tid >> 6;       // 0..3
  const int m  = M0 + r;
  const int bb = m >> 11;       // / 2048
  const int nn = m & 2047;
  const float* arow = x + (size_t)bb * (Hn * En) + (size_t)nn * En;

  // B staging: transpose conv_w[c][k] -> sB[k][c]
  const int cl = tid >> 3;         // 0..31 local col
  const int kq = (tid & 7) * 8;    // 0..56
  const float* brow = cw + (size_t)(N0 + cl) * KCONV;

  v8f acc = {};
  for (int kb = 0; kb < KCONV; kb += BK) {
    // stage A (64 x 64), rows contiguous in global memory
    {
      const float4* g = (const float4*)(arow + kb + q * 16);
      float4* s4 = (float4*)(&sA[r * LDA + q * 16]);
      s4[0] = g[0]; s4[1] = g[1]; s4[2] = g[2]; s4[3] = g[3];
    }
    // stage B transposed (64 x 32)
    {
      float4 w0 = *(const float4*)(brow + kb + kq);
      float4 w1 = *(const float4*)(brow + kb + kq + 4);
      sB[(kq + 0) * LDB + cl] = w0.x;
      sB[(kq + 1) * LDB + cl] = w0.y;
      sB[(kq + 2) * LDB + cl] = w0.z;
      sB[(kq + 3) * LDB + cl] = w0.w;
      sB[(kq + 4) * LDB + cl] = w1.x;
      sB[(kq + 5) * LDB + cl] = w1.y;
      sB[(kq + 6) * LDB + cl] = w1.z;
      sB[(kq + 7) * LDB + cl] = w1.w;
    }
    __syncthreads();

    const int ar = wm * 16 + (lane & 15);
    const int kh = (lane >> 4) * 2;        // 0 or 2
    const int bc = wn * 16 + (lane & 15);
#pragma unroll
    for (int kk = 0; kk < BK; kk += 4) {
      v2f a, b;
      a[0] = sA[ar * LDA + kk + kh + 0];
      a[1] = sA[ar * LDA + kk + kh + 1];
      b[0] = sB[(kk + kh + 0) * LDB + bc];
      b[1] = sB[(kk + kh + 1) * LDB + bc];
      acc = __builtin_amdgcn_wmma_f32_16x16x4_f32(false, a, false, b,
                                                  (short)0, acc, false, false);
    }
    __syncthreads();
  }

  // epilogue: bias + relu, D layout: VGPR v -> row v (lanes 0-15) / v+8 (16-31)
  const int colg  = N0 + wn * 16 + (lane & 15);
  const float bias = cb[colg];
  const int mrow0 = M0 + wm * 16 + ((lane < 16) ? 0 : 8);
#pragma unroll
  for (int v = 0; v < 8; ++v) {
    float val = acc[v] + bias;
    val = val > 0.0f ? val : 0.0f;
    out[(size_t)(mrow0 + v) * Cn + colg] = val;
  }
}

// ---------------- generic h@W GEMM (M=65536, K=128, N=128) ----------------
__global__ __launch_bounds__(256) void k_gemm_hw_cgx(const float* __restrict__ h,
                                                     const float* __restrict__ W,
                                                     float* __restrict__ out) {
  __shared__ float sA[BM * LDA];
  __shared__ float sB[BK * LDB];
  const int tid  = threadIdx.x;
  const int M0   = blockIdx.x * BM;
  const int N0   = blockIdx.y * BN;
  const int lane = tid & 31;
  const int wave = tid >> 5;
  const int wm   = wave >> 1;
  const int wn   = wave & 1;

  const int r = tid & 63;
  const int q = tid >> 6; // 0..3

  // B staging: W already K x N row-major
  const int kl = tid >> 2;        // 0..63
  const int cq = (tid & 3) * 8;   // 0,8,16,24

  v8f acc = {};
  for (int kb = 0; kb < 128; kb += BK) {
    {
      const float* ga = h + (size_t)(M0 + r) * 128 + kb + q * 16;
      float* la = &sA[r * LDA + q * 16];
#if USE_ASYNC
      async_copy16(ga + 0,  la + 0);
      async_copy16(ga + 4,  la + 4);
      async_copy16(ga + 8,  la + 8);
      async_copy16(ga + 12, la + 12);
#else
      const float4* g = (const float4*)ga;
      float4* s4 = (float4*)la;
      s4[0] = g[0]; s4[1] = g[1]; s4[2] = g[2]; s4[3] = g[3];
#endif
    }
    {
      const float* wr = W + (size_t)(kb + kl) * 128 + N0 + cq;
      float4 w0 = *(const float4*)(wr);
      float4 w1 = *(const float4*)(wr + 4);
      sB[kl * LDB + cq + 0] = w0.x;
      sB[kl * LDB + cq + 1] = w0.y;
      sB[kl * LDB + cq + 2] = w0.z;
      sB[kl * LDB + cq + 3] = w0.w;
      sB[kl * LDB + cq + 4] = w1.x;
      sB[kl * LDB + cq + 5] = w1.y;
      sB[kl * LDB + cq + 6] = w1.z;
      sB[kl * LDB + cq + 7] = w1.w;
    }
#if USE_ASYNC
    async_wait0();
#endif
    __syncthreads();

    const int ar = wm * 16 + (lane & 15);
    const int kh = (lane >> 4) * 2;
    const int bc = wn * 16 + (lane & 15);
#pragma unroll
    for (int kk = 0; kk < BK; kk += 4) {
      v2f a, b;
      a[0] = sA[ar * LDA + kk + kh + 0];
      a[1] = sA[ar * LDA + kk + kh + 1];
      b[0] = sB[(kk + kh + 0) * LDB + bc];
      b[1] = sB[(kk + kh + 1) * LDB + bc];
      acc = __builtin_amdgcn_wmma_f32_16x16x4_f32(false, a, false, b,
                                                  (short)0, acc, false, false);
    }
    __syncthreads();
  }

  const int colg  = N0 + wn * 16 + (lane & 15);
  const int mrow0 = M0 + wm * 16 + ((lane < 16) ? 0 : 8);
#pragma unroll
  for (int v = 0; v < 8; ++v) {
    out[(size_t)(mrow0 + v) * Cn + colg] = acc[v]; // no bias/relu here
  }
}

// ---------------- GCN aggregation ----------------
// init: self-loop term + bias: out = hw * dinv[n]^2 + bias[c]
__global__ void k_agg_init_cgx(const float* __restrict__ hw,
                               const float* __restrict__ dinv,
                               const float* __restrict__ bias,
                               float* __restrict__ out) {
  int i4 = blockIdx.x * blockDim.x + threadIdx.x;   // float4 index
  if (i4 < HNC / 4) {
    int base = i4 * 4;
    int n = (base >> 7) & 2047;
    int c = base & 127;
    float dn = dinv[n];
    float w = dn * dn;
    float4 hv = ((const float4*)hw)[i4];
    float4 bv = *(const float4*)(bias + c);
    float4 o;
    o.x = hv.x * w + bv.x;
    o.y = hv.y * w + bv.y;
    o.z = hv.z * w + bv.z;
    o.w = hv.w * w + bv.w;
    ((float4*)out)[i4] = o;
  }
}

// edge scatter: out[b, dst, c] += hw[b, src, c] * norm[e]
__global__ __launch_bounds__(256) void k_agg_edges_cgx(const long long* __restrict__ ei,
                                                       const float* __restrict__ norm,
                                                       const float* __restrict__ hw,
                                                       float* __restrict__ out) {
  const int e = blockIdx.x;
  const int s = (int)ei[e];
  const int d = (int)ei[NEn + e];
  const float w = norm[e];
  for (int i = threadIdx.x; i < Bn * Cn; i += 256) {
    int b = i >> 7;
    int c = i & 127;
    float v = hw[((size_t)b * Nn + s) * Cn + c] * w;
    atomicAdd(&out[((size_t)b * Nn + d) * Cn + c], v);
  }
}

__global__ void k_relu_cgx(float* h) {
  int i4 = blockIdx.x * blockDim.x + threadIdx.x;
  if (i4 < HNC / 4) {
    float4 v = ((float4*)h)[i4];
    v.x = v.x > 0.0f ? v.x : 0.0f;
    v.y = v.y > 0.0f ? v.y : 0.0f;
    v.z = v.z > 0.0f ? v.z : 0.0f;
    v.w = v.w > 0.0f ? v.w : 0.0f;
    ((float4*)h)[i4] = v;
  }
}

// ---------------- MLP head ----------------
// split-K over K=262144; feat pre-initialized to b1; atomicAdd partials.
__global__ __launch_bounds__(256) void k_mlp1_cgx(const float* __restrict__ h2,
                                                  const float* __restrict__ W1,
                                                  float* __restrict__ feat) {
  const int KS2 = 1024;                           // per-block K slice
  const int kb = blockIdx.x * KS2;
  const int b  = threadIdx.x & 31;
  const int jg = threadIdx.x >> 5;                // 0..7
  const float* hb = h2 + (size_t)b * (Nn * Cn) + kb;
  for (int j = jg; j < MLPn; j += 8) {
    const float* wj = W1 + (size_t)kb * MLPn + j;
    __builtin_prefetch(wj, 0, 1);
    float a0 = 0.0f, a1 = 0.0f;
    for (int k = 0; k < KS2; k += 2) {
      a0 += hb[k] * wj[(size_t)k * MLPn];
      a1 += hb[k + 1] * wj[(size_t)(k + 1) * MLPn];
    }
    atomicAdd(&feat[b * MLPn + j], a0 + a1);
  }
}

__global__ void k_final_cgx(const float* __restrict__ feat,
                            const float* __restrict__ W2,
                            const float* __restrict__ b2,
                            float* __restrict__ out) {
  int b = threadIdx.x;
  if (b < Bn) {
    float acc = b2[0];
    for (int j = 0; j < MLPn; ++j) {
      float f = feat[b * MLPn + j];
      f = f > 0.0f ? f : 0.0f;
      acc += f * W2[j];
    }
    out[b] = acc;
  }
}

// ---------------- launcher ----------------
extern "C" void kernel_launch(void* const* d_in, const int* in_sizes, int n_in,
                              void* d_out, int out_size, void* d_ws, size_t ws_size,
                              hipStream_t stream) {
  const float*     x      = (const float*)d_in[0];
  const long long* ei     = (const long long*)d_in[1];
  const float*     conv_w = (const float*)d_in[2];
  const float*     conv_b = (const float*)d_in[3];
  const float*     gW1    = (const float*)d_in[4];
  const float*     gb1    = (const float*)d_in[5];
  const float*     gW2    = (const float*)d_in[6];
  const float*     gb2    = (const float*)d_in[7];
  const float*     W1     = (const float*)d_in[8];
  const float*     b1     = (const float*)d_in[9];
  const float*     W2     = (const float*)d_in[10];
  const float*     b2     = (const float*)d_in[11];
  (void)in_sizes; (void)n_in; (void)out_size; (void)ws_size;

  float* ws   = (float*)d_ws;
  float* hA   = ws;                       // 8388608 floats
  float* hB   = ws + (size_t)HNC;         // 8388608 floats
  float* deg  = ws + (size_t)2 * HNC;     // 2048 (deg -> dinv in place)
  float* norm = deg + Nn;                 // 32768
  float* feat = norm + NEn;               // 3200

  // norm precompute
  k_init_cgx<<<32, 256, 0, stream>>>(deg, feat, b1);
  k_deg_cgx<<<NEn / 256, 256, 0, stream>>>(ei, deg);
  k_dinv_cgx<<<(Nn + 255) / 256, 256, 0, stream>>>(deg);
  k_norm_cgx<<<NEn / 256, 256, 0, stream>>>(ei, deg, norm);

  dim3 ggrid(MROWS / BM, Cn / BN);        // (1024, 4)
  const int ew4 = (HNC / 4 + 255) / 256;  // elementwise grid (8192)

  // conv -> hA (bias + relu fused)
  k_conv_gemm_cgx<<<ggrid, 256, 0, stream>>>(x, conv_w, conv_b, hA);

  // GCN layer 1
  k_gemm_hw_cgx<<<ggrid, 256, 0, stream>>>(hA, gW1, hB);
  k_agg_init_cgx<<<ew4, 256, 0, stream>>>(hB, deg, gb1, hA);
  k_agg_edges_cgx<<<NEn, 256, 0, stream>>>(ei, norm, hB, hA);
  k_relu_cgx<<<ew4, 256, 0, stream>>>(hA);

  // GCN layer 2
  k_gemm_hw_cgx<<<ggrid, 256, 0, stream>>>(hA, gW2, hB);
  k_agg_init_cgx<<<ew4, 256, 0, stream>>>(hB, deg, gb2, hA);
  k_agg_edges_cgx<<<NEn, 256, 0, stream>>>(ei, norm, hB, hA);
  k_relu_cgx<<<ew4, 256, 0, stream>>>(hA);

  // MLP head
  k_mlp1_cgx<<<(Nn * Cn) / 1024, 256, 0, stream>>>(hA, W1, feat); // 256 blocks
  k_final_cgx<<<1, 32, 0, stream>>>(feat, W2, b2, (float*)d_out);
}